// GCN_24592982737081
// MI455X (gfx1250) — compile-verified
//
#include <hip/hip_runtime.h>

#define N_NODES  100000
#define N_EDGES  1600000
#define D        64
#define D_OUT    10
#define N_GRAPHS 512

typedef __attribute__((ext_vector_type(2))) float v2f;
typedef __attribute__((ext_vector_type(8))) float v8f;

// ---------------------------------------------------------------- fill zero
__global__ void gcn_fill_zero(float* __restrict__ p, int n) {
    int i = blockIdx.x * blockDim.x + threadIdx.x;
    if (i < n) p[i] = 0.0f;
}

// ------------------------------------------------- edge scatter-add (segment_sum)
// 16 lanes per edge, each lane moves a float4 (one 256B row per edge total).
__global__ void gcn_scatter_add(const float* __restrict__ h,
                                const int* __restrict__ src,
                                const int* __restrict__ dst,
                                float* __restrict__ agg) {
    long long tid = (long long)blockIdx.x * blockDim.x + threadIdx.x;
    int edge = (int)(tid >> 4);
    int f4   = (int)(tid & 15);
    if (edge >= N_EDGES) return;
    int s = src[edge];
    int d = dst[edge];
    const float4 v = *(const float4*)&h[(long long)s * D + f4 * 4];
    float* o = &agg[(long long)d * D + f4 * 4];
    atomicAdd(o + 0, v.x);
    atomicAdd(o + 1, v.y);
    atomicAdd(o + 2, v.z);
    atomicAdd(o + 3, v.w);
}

// ---------------------------------------------- fused dual-GEMM layer via WMMA
// out = [relu]( A1 @ W1 + bias + A2 @ W2 ),  A* : [N_NODES x 64], W* : [64 x 64]
// One wave computes one 16x16 output tile with V_WMMA_F32_16X16X4_F32.
__global__ void gcn_layer_wmma(const float* __restrict__ A1, const float* __restrict__ W1,
                               const float* __restrict__ A2, const float* __restrict__ W2,
                               const float* __restrict__ bias,
                               float* __restrict__ out, int relu) {
    int wave = (blockIdx.x * blockDim.x + threadIdx.x) >> 5;
    int lane = threadIdx.x & 31;
    const int tilesN = D / 16;                 // 4
    int tM = wave / tilesN;
    int tN = wave - tM * tilesN;
    int m0 = tM * 16;
    int n0 = tN * 16;
    if (m0 >= N_NODES) return;                 // wave-uniform: EXEC stays all-1s

    int lid = lane & 15;                       // row (A) / col (B,C,D) within tile
    int hs  = lane >> 4;                       // half-select

    float bv = bias[n0 + lid];
    v8f acc = {bv, bv, bv, bv, bv, bv, bv, bv};

    const float* a1p = &A1[(long long)(m0 + lid) * D + 2 * hs];
    const float* a2p = &A2[(long long)(m0 + lid) * D + 2 * hs];

#pragma unroll
    for (int k0 = 0; k0 < D; k0 += 4) {
        // A fragment: lane holds row m0+lid, K = k0+2*hs, k0+2*hs+1 (contiguous)
        v2f a1 = *(const v2f*)(a1p + k0);
        v2f b1;
        b1.x = W1[(k0 + 2 * hs)     * D + n0 + lid];
        b1.y = W1[(k0 + 2 * hs + 1) * D + n0 + lid];
        acc = __builtin_amdgcn_wmma_f32_16x16x4_f32(false, a1, false, b1,
                                                    (short)0, acc, false, false);

        v2f a2 = *(const v2f*)(a2p + k0);
        v2f b2;
        b2.x = W2[(k0 + 2 * hs)     * D + n0 + lid];
        b2.y = W2[(k0 + 2 * hs + 1) * D + n0 + lid];
        acc = __builtin_amdgcn_wmma_f32_16x16x4_f32(false, a2, false, b2,
                                                    (short)0, acc, false, false);
    }

#pragma unroll
    for (int r = 0; r < 8; r++) {
        int m = m0 + r + 8 * hs;               // C/D layout: VGPR r -> row r (+8 for hi half)
        float v = acc[r];
        if (relu) v = fmaxf(v, 0.0f);
        out[(long long)m * D + n0 + lid] = v;
    }
}

// ----------------------------------------------------- mean-pool accumulation
__global__ void gcn_pool_accum(const float* __restrict__ h, const int* __restrict__ batch,
                               float* __restrict__ sums, float* __restrict__ counts) {
    long long tid = (long long)blockIdx.x * blockDim.x + threadIdx.x;
    int node = (int)(tid >> 6);
    int f    = (int)(tid & 63);
    if (node >= N_NODES) return;
    int g = batch[node];
    atomicAdd(&sums[g * D + f], h[(long long)node * D + f]);
    if (f == 0) atomicAdd(&counts[g], 1.0f);
}

// ----------------------------------------------------- final linear (512x10)
__global__ void gcn_final_linear(const float* __restrict__ sums, const float* __restrict__ counts,
                                 const float* __restrict__ Wlin, const float* __restrict__ blin,
                                 float* __restrict__ out) {
    int tid = blockIdx.x * blockDim.x + threadIdx.x;
    if (tid >= N_GRAPHS * D_OUT) return;
    int g = tid / D_OUT;
    int j = tid - g * D_OUT;
    float c = counts[g];
    c = (c > 1.0f) ? c : 1.0f;
    float inv = 1.0f / c;
    float acc = blin[j];
#pragma unroll 8
    for (int k = 0; k < D; k++)
        acc += sums[g * D + k] * inv * Wlin[k * D_OUT + j];
    out[tid] = acc;
}

extern "C" void kernel_launch(void* const* d_in, const int* in_sizes, int n_in,
                              void* d_out, int out_size, void* d_ws, size_t ws_size,
                              hipStream_t stream) {
    const float* x      = (const float*)d_in[0];
    const int*   ei     = (const int*)d_in[1];
    const int*   batch  = (const int*)d_in[2];
    const float* Wrel1  = (const float*)d_in[3];
    const float* brel1  = (const float*)d_in[4];
    const float* Wroot1 = (const float*)d_in[5];
    const float* Wrel2  = (const float*)d_in[6];
    const float* brel2  = (const float*)d_in[7];
    const float* Wroot2 = (const float*)d_in[8];
    const float* Wrel3  = (const float*)d_in[9];
    const float* brel3  = (const float*)d_in[10];
    const float* Wroot3 = (const float*)d_in[11];
    const float* Wlin   = (const float*)d_in[12];
    const float* blin   = (const float*)d_in[13];
    float* out = (float*)d_out;

    // workspace carve-out (~77 MB)
    float* agg    = (float*)d_ws;
    float* h_a    = agg  + (size_t)N_NODES * D;
    float* h_b    = h_a  + (size_t)N_NODES * D;
    float* sums   = h_b  + (size_t)N_NODES * D;
    float* counts = sums + (size_t)N_GRAPHS * D;

    const int* src = ei;
    const int* dst = ei + N_EDGES;

    const int nAgg          = N_NODES * D;
    const int fillBlocks    = (nAgg + 255) / 256;
    const int scatterBlocks = (N_EDGES * 16 + 255) / 256;
    const int gemmWaves     = ((N_NODES + 15) / 16) * (D / 16);   // 25000
    const int gemmBlocks    = (gemmWaves * 32 + 255) / 256;       // 3125

    // ---- layer 1: h_a = relu(seg_sum(x) @ Wrel1 + brel1 + x @ Wroot1)
    gcn_fill_zero<<<fillBlocks, 256, 0, stream>>>(agg, nAgg);
    gcn_scatter_add<<<scatterBlocks, 256, 0, stream>>>(x, src, dst, agg);
    gcn_layer_wmma<<<gemmBlocks, 256, 0, stream>>>(agg, Wrel1, x, Wroot1, brel1, h_a, 1);

    // ---- layer 2: h_b = relu(seg_sum(h_a) @ Wrel2 + brel2 + h_a @ Wroot2)
    gcn_fill_zero<<<fillBlocks, 256, 0, stream>>>(agg, nAgg);
    gcn_scatter_add<<<scatterBlocks, 256, 0, stream>>>(h_a, src, dst, agg);
    gcn_layer_wmma<<<gemmBlocks, 256, 0, stream>>>(agg, Wrel2, h_a, Wroot2, brel2, h_b, 1);

    // ---- layer 3 (no relu): h_a = seg_sum(h_b) @ Wrel3 + brel3 + h_b @ Wroot3
    gcn_fill_zero<<<fillBlocks, 256, 0, stream>>>(agg, nAgg);
    gcn_scatter_add<<<scatterBlocks, 256, 0, stream>>>(h_b, src, dst, agg);
    gcn_layer_wmma<<<gemmBlocks, 256, 0, stream>>>(agg, Wrel3, h_b, Wroot3, brel3, h_a, 0);

    // ---- mean pool + final linear
    const int nPoolZero = N_GRAPHS * D + N_GRAPHS;
    gcn_fill_zero<<<(nPoolZero + 255) / 256, 256, 0, stream>>>(sums, nPoolZero);
    gcn_pool_accum<<<((N_NODES * 64) + 255) / 256, 256, 0, stream>>>(h_a, batch, sums, counts);
    gcn_final_linear<<<(N_GRAPHS * D_OUT + 255) / 256, 256, 0, stream>>>(sums, counts, Wlin, blin, out);
}